// GNN_44306882625625
// MI455X (gfx1250) — compile-verified
//
#include <hip/hip_runtime.h>

// ---------------------------------------------------------------------------
// 2-layer GraphSAGE for MI455X (gfx1250, wave32).
// Layer-1 dense part uses V_WMMA_F32_16X16X4_F32 (full f32 precision; problem
// is L2/atomic-bandwidth bound on the edge scatter, so fp32 WMMA is the right
// precision/perf point). Layer 2 is algebraically collapsed to column sums.
// ---------------------------------------------------------------------------

typedef float v2f __attribute__((ext_vector_type(2)));
typedef float v8f __attribute__((ext_vector_type(8)));

#define N_NODES 50000
#define IN_FEAT 64
#define HIDDEN  128
#define OUT_F   10

// ---- degree count: cnt[dst] += 1 --------------------------------------------
__global__ void gnn_count(const int* __restrict__ dst, float* __restrict__ cnt, int E) {
    int e = blockIdx.x * blockDim.x + threadIdx.x;
    if (e < E) atomicAdd(&cnt[dst[e]], 1.0f);
}

// ---- cnt[n] <- 1 / max(cnt[n], 1) ------------------------------------------
__global__ void gnn_rcnt(float* __restrict__ cnt, int N) {
    int n = blockIdx.x * blockDim.x + threadIdx.x;
    if (n < N) cnt[n] = 1.0f / fmaxf(cnt[n], 1.0f);
}

// ---- edge scatter: agg[dst*F + f] += feat[src*F + f], float4 per thread -----
template <int F4>   // number of float4 chunks per row (16 for 64-wide, 32 for 128-wide)
__global__ void gnn_scatter(const int* __restrict__ src, const int* __restrict__ dst,
                            const float4* __restrict__ feat, float* __restrict__ agg, int E) {
    int tid = blockIdx.x * blockDim.x + threadIdx.x;
    if (tid >= E * F4) return;
    int e = tid / F4;
    int q = tid % F4;
    int s = src[e], d = dst[e];
    float4 v = feat[(size_t)s * F4 + q];
    float* p = agg + (size_t)d * (F4 * 4) + q * 4;
    atomicAdd(p + 0, v.x);
    atomicAdd(p + 1, v.y);
    atomicAdd(p + 2, v.z);
    atomicAdd(p + 3, v.w);
}

// ---- layer-1 GEMM via WMMA f32 16x16x4 --------------------------------------
// h[n,j] = relu( (agg1[n,:]*rcnt[n]) . W1l[j,:] + b1[j] + x[n,:] . W1r[j,:] )
// Block = 256 threads = 8 waves; wave w handles hidden tile [16w,16w+16),
// block b handles node tile [16b, 16b+16). 50000/16 = 3125 blocks exactly.
__global__ __launch_bounds__(256) void gnn_gemm1_wmma(
        const float* __restrict__ agg1, const float* __restrict__ rcnt,
        const float* __restrict__ x,
        const float* __restrict__ W1l, const float* __restrict__ b1,
        const float* __restrict__ W1r, float* __restrict__ h) {
    const int lane = threadIdx.x & 31;
    const int wave = threadIdx.x >> 5;
    const int nb = blockIdx.x << 4;     // node-tile base
    const int hb = wave << 4;           // hidden-tile base
    const int m  = lane & 15;           // row within tile (A: M, B: N)
    const int kh = (lane >> 4) << 1;    // K sub-offset: 0 (lanes 0-15) or 2 (lanes 16-31)

    const float rc = rcnt[nb + m];

    const float* Arow = agg1 + (size_t)(nb + m) * IN_FEAT + kh;
    const float* Xrow = x    + (size_t)(nb + m) * IN_FEAT + kh;
    const float* Bl   = W1l  + (size_t)(hb + m) * IN_FEAT + kh;  // B[k][n] = W[n][k]
    const float* Br   = W1r  + (size_t)(hb + m) * IN_FEAT + kh;

    v8f acc = {};

    // neighbor-mean path: (agg1 * rcnt) @ W1l^T
#pragma unroll
    for (int kk = 0; kk < IN_FEAT; kk += 4) {
        v2f a; a.x = Arow[kk] * rc; a.y = Arow[kk + 1] * rc;
        v2f b; b.x = Bl[kk];        b.y = Bl[kk + 1];
        acc = __builtin_amdgcn_wmma_f32_16x16x4_f32(
            /*neg_a=*/false, a, /*neg_b=*/false, b,
            /*c_mod=*/(short)0, acc, /*reuse_a=*/false, /*reuse_b=*/false);
    }
    // root path: x @ W1r^T
#pragma unroll
    for (int kk = 0; kk < IN_FEAT; kk += 4) {
        v2f a; a.x = Xrow[kk]; a.y = Xrow[kk + 1];
        v2f b; b.x = Br[kk];   b.y = Br[kk + 1];
        acc = __builtin_amdgcn_wmma_f32_16x16x4_f32(
            false, a, false, b, (short)0, acc, false, false);
    }

    // C/D layout: VGPR r -> row (r + 8*(lane>=16)), col = lane&15
    const int col   = lane & 15;
    const int rbase = (lane >> 4) << 3;
    const float bias = b1[hb + col];
#pragma unroll
    for (int r = 0; r < 8; ++r) {
        float v = acc[r] + bias;
        v = v > 0.0f ? v : 0.0f;   // ReLU
        h[(size_t)(nb + rbase + r) * HIDDEN + hb + col] = v;
    }
}

// ---- fused column sums: S1 = sum_n agg2[n,:]*rcnt[n] ; S2 = sum_n h[n,:] ----
__global__ __launch_bounds__(128) void gnn_colsum(
        const float* __restrict__ agg2, const float* __restrict__ h,
        const float* __restrict__ rcnt,
        float* __restrict__ S1, float* __restrict__ S2, int N) {
    const int f  = threadIdx.x;            // 0..127
    const int n0 = blockIdx.x * 256;
    const int n1 = min(n0 + 256, N);
    float s1 = 0.0f, s2 = 0.0f;
    for (int n = n0; n < n1; ++n) {
        float rc = rcnt[n];
        s1 += agg2[(size_t)n * HIDDEN + f] * rc;
        s2 += h   [(size_t)n * HIDDEN + f];
    }
    atomicAdd(&S1[f], s1);
    atomicAdd(&S2[f], s2);
}

// ---- out[o] = S1 . W2l[o,:] + N*b2[o] + S2 . W2r[o,:] -----------------------
__global__ void gnn_final(const float* __restrict__ S1, const float* __restrict__ S2,
                          const float* __restrict__ W2l, const float* __restrict__ b2,
                          const float* __restrict__ W2r, float* __restrict__ out, int N) {
    int o = threadIdx.x;
    if (o >= OUT_F) return;
    float acc = (float)N * b2[o];
    for (int k = 0; k < HIDDEN; ++k)
        acc += S1[k] * W2l[o * HIDDEN + k] + S2[k] * W2r[o * HIDDEN + k];
    out[o] = acc;
}

extern "C" void kernel_launch(void* const* d_in, const int* in_sizes, int n_in,
                              void* d_out, int out_size, void* d_ws, size_t ws_size,
                              hipStream_t stream) {
    // Inputs per reference setup_inputs() order
    const float* x   = (const float*)d_in[0];        // [N, 64]
    const int*   ei  = (const int*)  d_in[1];        // [2, E]
    const float* W1l = (const float*)d_in[2];        // [128, 64]
    const float* b1  = (const float*)d_in[3];        // [128]
    const float* W1r = (const float*)d_in[4];        // [128, 64]
    const float* W2l = (const float*)d_in[5];        // [10, 128]
    const float* b2  = (const float*)d_in[6];        // [10]
    const float* W2r = (const float*)d_in[7];        // [10, 128]
    float*       out = (float*)d_out;

    const int N = in_sizes[0] / IN_FEAT;             // 50000
    const int E = in_sizes[1] / 2;                   // 800000
    const int* src = ei;
    const int* dst = ei + E;

    // Workspace partition (floats): cnt[N] | agg1[N*64] | h[N*128] | agg2[N*128] | S1[128] | S2[128]
    float* ws   = (float*)d_ws;
    float* cnt  = ws;
    float* agg1 = cnt  + N;
    float* h    = agg1 + (size_t)N * IN_FEAT;
    float* agg2 = h    + (size_t)N * HIDDEN;
    float* S1   = agg2 + (size_t)N * HIDDEN;
    float* S2   = S1 + HIDDEN;

    // Zero the accumulation buffers (graph-capture-safe async memsets)
    hipMemsetAsync(cnt,  0, (size_t)N * sizeof(float), stream);
    hipMemsetAsync(agg1, 0, (size_t)N * IN_FEAT * sizeof(float), stream);
    hipMemsetAsync(agg2, 0, (size_t)N * HIDDEN * sizeof(float), stream);
    hipMemsetAsync(S1,   0, 2 * HIDDEN * sizeof(float), stream);

    // 1) degree counts
    gnn_count<<<(E + 255) / 256, 256, 0, stream>>>(dst, cnt, E);
    // 2) layer-1 scatter: agg1[dst] += x[src]
    gnn_scatter<16><<<((size_t)E * 16 + 255) / 256, 256, 0, stream>>>(
        src, dst, (const float4*)x, agg1, E);
    // 3) reciprocal counts (in place)
    gnn_rcnt<<<(N + 255) / 256, 256, 0, stream>>>(cnt, N);
    // 4) layer-1 dense via WMMA: h = relu(mean @ W1l^T + b1 + x @ W1r^T)
    gnn_gemm1_wmma<<<N / 16, 256, 0, stream>>>(agg1, cnt, x, W1l, b1, W1r, h);
    // 5) layer-2 scatter: agg2[dst] += h[src]
    gnn_scatter<32><<<((size_t)E * 32 + 255) / 256, 256, 0, stream>>>(
        src, dst, (const float4*)h, agg2, E);
    // 6) column sums of mean2 and h
    gnn_colsum<<<(N + 255) / 256, 128, 0, stream>>>(agg2, h, cnt, S1, S2, N);
    // 7) final 10-wide output
    gnn_final<<<1, 32, 0, stream>>>(S1, S2, W2l, b2, W2r, out, N);
}